// YOLOLayer_9809705304728
// MI455X (gfx1250) — compile-verified
//
#include <hip/hip_runtime.h>
#include <math.h>

// ---------------- problem constants ----------------
#define FDIM 76
#define PDIM (FDIM * FDIM)      // 5776 pixels
#define CDIM 256                // input channels (K)
#define ODIM 255                // 3*(5+80) output channels
#define OPAD 256                // padded channel stride in workspace
#define BDIM 16                 // batch
#define NLAB 50                 // labels per batch
#define NCLS 80
#define IGNORE_THRE 0.7f
#define RECF 16                 // floats per label record

// GEMM tiling
#define TNPIX 64                // pixels per block (4 N-tiles)
#define XSTRIDE 80              // LDS row stride in floats (conflict-free: 16*k+col banks)
#define GWAVES 4                // waves per GEMM block
#define GTHREADS (GWAVES * 32)

// anchors / STRIDE(=8)
__device__ __constant__ float AW[9] = {1.25f, 2.0f,   4.125f, 3.75f, 7.75f,  7.375f, 14.5f,  19.5f,  46.625f};
__device__ __constant__ float AH[9] = {1.625f, 3.75f, 2.875f, 7.625f, 5.625f, 14.875f, 11.25f, 24.75f, 40.75f};

typedef float v2f __attribute__((ext_vector_type(2)));
typedef float v8f __attribute__((ext_vector_type(8)));

// =====================================================================
// Kernel 1: 1x1 conv as WMMA GEMM.  out[b][p][o] = sum_c xin[b][c][p]*W[o][c] + bias[o]
// Block: 128 threads = 4 waves; 64-pixel panel; each wave owns 4 M-tiles x 4 N-tiles
// -> 16 v_wmma per K-step vs 8 memory instructions (0.5 mem/WMMA).
// X panel staged with CDNA5 async-to-LDS DMA (ASYNCcnt), 80KB LDS.
// =====================================================================
__global__ void __launch_bounds__(GTHREADS) yolo_gemm_kernel(
    const float* __restrict__ xin, const float* __restrict__ W,
    const float* __restrict__ bias, float* __restrict__ out)
{
    __shared__ float Xs[CDIM * XSTRIDE];   // [c][pixel] stride 80 -> 80KB
    __shared__ float Bs[OPAD];             // bias staged (slot 255 = 0)

    const int b   = blockIdx.y;
    const int p0  = blockIdx.x * TNPIX;
    const int tid = threadIdx.x;

    // ---- stage bias ----
    for (int i = tid; i < OPAD; i += GTHREADS)
        Bs[i] = (i < ODIM) ? bias[i] : 0.0f;

    int validPix = PDIM - p0;
    if (validPix > TNPIX) validPix = TNPIX;

    if (validPix < TNPIX) {                // tail block only: zero-fill panel
        for (int i = tid; i < CDIM * XSTRIDE; i += GTHREADS) Xs[i] = 0.0f;
    }
    __syncthreads();                       // zeros visible before async writes land

    // ---- async DMA stage: 256 rows x validPix floats, b128 per lane ----
    const float* gbase = xin + (size_t)b * CDIM * PDIM + p0;
    if (validPix == TNPIX) {
        for (int f = tid; f < CDIM * 16; f += GTHREADS) {     // 4096 float4
            const int row = f >> 4;
            const int c4  = f & 15;
            const int voff    = row * (PDIM * 4) + c4 * 16;   // global byte offset
            const unsigned lo = (unsigned)(uintptr_t)&Xs[row * XSTRIDE + c4 * 4];
            asm volatile("global_load_async_to_lds_b128 %0, %1, %2"
                         :: "v"(lo), "v"(voff), "s"(gbase) : "memory");
        }
    } else {
        for (int f = tid; f < CDIM * 16; f += GTHREADS) {
            const int row = f >> 4;
            const int c4  = f & 15;
            if (c4 * 4 < validPix) {
                const int voff    = row * (PDIM * 4) + c4 * 16;
                const unsigned lo = (unsigned)(uintptr_t)&Xs[row * XSTRIDE + c4 * 4];
                asm volatile("global_load_async_to_lds_b128 %0, %1, %2"
                             :: "v"(lo), "v"(voff), "s"(gbase) : "memory");
            }
        }
    }
    asm volatile("s_wait_asynccnt 0" ::: "memory");
    __syncthreads();

    // ---- compute ----
    const int wave  = tid >> 5;            // 0..3 -> M tiles 4w..4w+3
    const int lane  = tid & 31;
    const int lhalf = lane & 15;
    const int khalf = lane >> 4;

    v8f acc[4][4] = {};                    // [mtile][ntile]

    const float* wp[4];
    #pragma unroll
    for (int mt = 0; mt < 4; ++mt) {
        int r = (wave * 4 + mt) * 16 + lhalf;
        if (r > ODIM - 1) r = ODIM - 1;    // clamp padding row (slot never read)
        wp[mt] = W + (size_t)r * CDIM + khalf * 2;
    }

    for (int k0 = 0; k0 < CDIM; k0 += 4) {
        const int kk = k0 + khalf * 2;
        v2f a[4];
        #pragma unroll
        for (int mt = 0; mt < 4; ++mt) a[mt] = *(const v2f*)(wp[mt] + k0);
        v2f bf[4];
        #pragma unroll
        for (int nt = 0; nt < 4; ++nt) {
            bf[nt].x = Xs[kk * XSTRIDE + nt * 16 + lhalf];
            bf[nt].y = Xs[(kk + 1) * XSTRIDE + nt * 16 + lhalf];
        }
        #pragma unroll
        for (int mt = 0; mt < 4; ++mt)
            #pragma unroll
            for (int nt = 0; nt < 4; ++nt)
                acc[mt][nt] = __builtin_amdgcn_wmma_f32_16x16x4_f32(
                    false, a[mt], false, bf[nt], (short)0, acc[mt][nt],
                    false, false);
    }

    // ---- store: [b][p][OPAD], 8 consecutive o-floats per lane per tile ----
    #pragma unroll
    for (int nt = 0; nt < 4; ++nt) {
        const int p = p0 + nt * 16 + lhalf;
        if (p < PDIM) {
            float* orow = out + ((size_t)b * PDIM + p) * OPAD;
            #pragma unroll
            for (int mt = 0; mt < 4; ++mt) {
                const int ob = (wave * 4 + mt) * 16 + khalf * 8;
                float t[8];
                #pragma unroll
                for (int r = 0; r < 8; ++r) t[r] = acc[mt][nt][r] + Bs[ob + r];
                *(float4*)(orow + ob)     = make_float4(t[0], t[1], t[2], t[3]);
                *(float4*)(orow + ob + 4) = make_float4(t[4], t[5], t[6], t[7]);
            }
        }
    }
}

// =====================================================================
// Kernel 2: per-label record prep (800 threads total)
// rec[idx*16]: 0=match 1=best_n 2=ii 3=jj 4=tx 5=ty 6=tw 7=th 8=valid
//              9=scale 10=dx 11=dy 12=log(tw/aw) 13=log(th/ah) 14=cls 15=pad
// =====================================================================
__global__ void __launch_bounds__(256) yolo_label_kernel(
    const float* __restrict__ labels, float* __restrict__ rec)
{
    const int idx = blockIdx.x * blockDim.x + threadIdx.x;
    if (idx >= BDIM * NLAB) return;
    const float* l = labels + (size_t)idx * 5;
    const float cls = l[0], lx = l[1], ly = l[2], lw = l[3], lh = l[4];
    const bool valid = (cls + lx + ly + lw + lh) > 0.0f;

    const float tx = lx * FDIM, ty = ly * FDIM;
    const float tw = lw * FDIM, th = lh * FDIM;

    float best = -1.0f; int bi = 0;
    #pragma unroll
    for (int n = 0; n < 9; ++n) {
        const float mw = fminf(tw, AW[n]);
        const float mh = fminf(th, AH[n]);
        const float inter = (mw > 0.0f && mh > 0.0f) ? mw * mh : 0.0f;
        const float iou = inter / (tw * th + AW[n] * AH[n] - inter);
        if (iou > best) { best = iou; bi = n; }
    }
    const int  bn    = bi % 3;
    const bool match = valid && (bi < 3);

    float* r = rec + (size_t)idx * RECF;
    r[0]  = match ? 1.0f : 0.0f;
    r[1]  = (float)bn;
    r[2]  = floorf(tx);
    r[3]  = match ? floorf(ty) : (float)FDIM;       // jj=F => dropped
    r[4]  = tx;  r[5] = ty;  r[6] = tw;  r[7] = th;
    r[8]  = valid ? 1.0f : 0.0f;
    r[9]  = sqrtf(2.0f - tw * th / (float)(FDIM * FDIM));
    r[10] = tx - floorf(tx);
    r[11] = ty - floorf(ty);
    r[12] = logf(tw / AW[bn] + 1e-16f);
    r[13] = logf(th / AH[bn] + 1e-16f);
    r[14] = cls;
    r[15] = 0.0f;
}

// =====================================================================
// Kernel 3: per-cell loss. grid = (ceil(P/256), A=3, B=16); 256 threads.
// partials[bid*5 + {xy,wh,obj,cls,l2}]
// =====================================================================
__device__ inline float clogf_(float v) { return logf(fmaxf(v, 1e-12f)); }
__device__ inline float bcef_(float o, float t) {
    return -(t * clogf_(o) + (1.0f - t) * clogf_(1.0f - o));
}
__device__ inline float sigmoidf_(float x) { return 1.0f / (1.0f + expf(-x)); }
__device__ inline float wave_red_(float v) {
    #pragma unroll
    for (int off = 16; off > 0; off >>= 1) v += __shfl_xor(v, off, 32);
    return v;
}

__global__ void __launch_bounds__(256) yolo_loss_kernel(
    const float* __restrict__ outBuf, const float* __restrict__ rec,
    float* __restrict__ partials)
{
    __shared__ float recs[NLAB * RECF];
    __shared__ float wsum[8][5];

    const int b = blockIdx.z;
    const int a = blockIdx.y;
    const int p = blockIdx.x * 256 + threadIdx.x;

    for (int i = threadIdx.x; i < NLAB * RECF; i += 256)
        recs[i] = rec[(size_t)b * NLAB * RECF + i];
    __syncthreads();

    float sxy = 0.f, swh = 0.f, sob = 0.f, scl = 0.f, sl2 = 0.f;

    if (p < PDIM) {
        const int x = p % FDIM;
        const int y = p / FDIM;
        const float* ch = outBuf + ((size_t)b * PDIM + p) * OPAD + a * 85;

        const float sx = sigmoidf_(ch[0]);
        const float sy = sigmoidf_(ch[1]);
        const float rw = ch[2], rh = ch[3];
        const float so = sigmoidf_(ch[4]);
        const float pw = expf(rw) * AW[a];
        const float ph = expf(rh) * AH[a];
        const float px = sx + (float)x;
        const float py = sy + (float)y;

        float maxiou = 0.0f;
        int mrec = -1;
        for (int k = 0; k < NLAB; ++k) {
            const float* r = &recs[k * RECF];
            if (r[8] > 0.5f) {
                const float tx = r[4], ty = r[5], tw = r[6], th = r[7];
                const float tlx = fmaxf(px - 0.5f * pw, tx - 0.5f * tw);
                const float tly = fmaxf(py - 0.5f * ph, ty - 0.5f * th);
                const float brx = fminf(px + 0.5f * pw, tx + 0.5f * tw);
                const float bry = fminf(py + 0.5f * ph, ty + 0.5f * th);
                const float inter = (tlx < brx && tly < bry)
                                        ? (brx - tlx) * (bry - tly) : 0.0f;
                const float iou = inter / (pw * ph + tw * th - inter);
                maxiou = fmaxf(maxiou, iou);
            }
            if (r[0] > 0.5f && (int)r[1] == a && (int)r[2] == x && (int)r[3] == y)
                mrec = k;   // last matching label wins (scatter overwrite)
        }

        float objm = (maxiou > IGNORE_THRE) ? 0.0f : 1.0f;
        float t_ob = 0.0f;
        if (mrec >= 0) { objm = 1.0f; t_ob = 1.0f; }

        const float oo = so * objm;
        sob += bcef_(oo, t_ob);
        sl2 += (oo - t_ob) * (oo - t_ob);

        if (mrec >= 0) {
            const float* r = &recs[mrec * RECF];
            const float sc = r[9];
            const float w2 = sc * sc;
            sxy += w2 * (bcef_(sx, r[10]) + bcef_(sy, r[11]));
            sl2 += (sx - r[10]) * (sx - r[10]) + (sy - r[11]) * (sy - r[11]);
            const float dw = (rw - r[12]) * sc;
            const float dh = (rh - r[13]) * sc;
            swh += 0.5f * (dw * dw + dh * dh);
            sl2 += dw * dw + dh * dh;
            const int ci = (int)r[14];
            for (int c = 0; c < NCLS; ++c) {
                const float sv = sigmoidf_(ch[5 + c]);
                const float tv = (c == ci) ? 1.0f : 0.0f;
                scl += bcef_(sv, tv);
                sl2 += (sv - tv) * (sv - tv);
            }
        }
    }

    float vals[5] = {sxy, swh, sob, scl, sl2};
    const int wave = threadIdx.x >> 5, lane = threadIdx.x & 31;
    #pragma unroll
    for (int j = 0; j < 5; ++j) {
        const float rsum = wave_red_(vals[j]);
        if (lane == 0) wsum[wave][j] = rsum;
    }
    __syncthreads();
    if (threadIdx.x == 0) {
        const int bid = blockIdx.x + gridDim.x * (blockIdx.y + gridDim.y * blockIdx.z);
        #pragma unroll
        for (int j = 0; j < 5; ++j) {
            float s = 0.f;
            for (int w = 0; w < 8; ++w) s += wsum[w][j];
            partials[(size_t)bid * 5 + j] = s;
        }
    }
}

// =====================================================================
// Kernel 4: deterministic final reduction -> 6 outputs
// =====================================================================
__global__ void __launch_bounds__(256) yolo_final_kernel(
    const float* __restrict__ partials, int n, float* __restrict__ out)
{
    __shared__ float wsum[8][5];
    const int tid = threadIdx.x;
    float s[5] = {0.f, 0.f, 0.f, 0.f, 0.f};
    for (int i = tid; i < n; i += 256)
        #pragma unroll
        for (int j = 0; j < 5; ++j) s[j] += partials[(size_t)i * 5 + j];

    const int wave = tid >> 5, lane = tid & 31;
    #pragma unroll
    for (int j = 0; j < 5; ++j) {
        const float r = wave_red_(s[j]);
        if (lane == 0) wsum[wave][j] = r;
    }
    __syncthreads();
    if (tid == 0) {
        float t[5];
        #pragma unroll
        for (int j = 0; j < 5; ++j) {
            float acc = 0.f;
            for (int w = 0; w < 8; ++w) acc += wsum[w][j];
            t[j] = acc;
        }
        out[0] = t[0] + t[1] + t[2] + t[3];   // loss
        out[1] = t[0];                        // loss_xy
        out[2] = t[1];                        // loss_wh
        out[3] = t[2];                        // loss_obj
        out[4] = t[3];                        // loss_cls
        out[5] = t[4];                        // loss_l2
    }
}

// =====================================================================
extern "C" void kernel_launch(void* const* d_in, const int* in_sizes, int n_in,
                              void* d_out, int out_size, void* d_ws, size_t ws_size,
                              hipStream_t stream)
{
    const float* xin    = (const float*)d_in[0];   // [16,256,76,76]
    const float* labels = (const float*)d_in[1];   // [16,50,5]
    const float* conv_w = (const float*)d_in[2];   // [255,256]
    const float* conv_b = (const float*)d_in[3];   // [255]

    float* outBuf   = (float*)d_ws;                                   // B*P*OPAD floats
    float* rec      = outBuf + (size_t)BDIM * PDIM * OPAD;            // 800*16 floats
    float* partials = rec + (size_t)BDIM * NLAB * RECF;               // nblk*5 floats

    const int ngx = (PDIM + TNPIX - 1) / TNPIX;  // 91 pixel panels
    yolo_gemm_kernel<<<dim3(ngx, BDIM), GTHREADS, 0, stream>>>(
        xin, conv_w, conv_b, outBuf);

    yolo_label_kernel<<<(BDIM * NLAB + 255) / 256, 256, 0, stream>>>(labels, rec);

    const int pblk = (PDIM + 255) / 256;     // 23
    dim3 lgrid(pblk, 3, BDIM);               // 23 x 3 x 16 = 1104 blocks
    yolo_loss_kernel<<<lgrid, 256, 0, stream>>>(outBuf, rec, partials);

    yolo_final_kernel<<<1, 256, 0, stream>>>(partials, pblk * 3 * BDIM,
                                             (float*)d_out);
}